// Net_16690242912864
// MI455X (gfx1250) — compile-verified
//
#include <hip/hip_runtime.h>

#define HID      128
#define NNODES   50000
#define NEDGES   800000
#define NGRAPHS  64
#define NLAYERS  2
#define BN_EPS   1e-5f
#define ROWBLKS  (NNODES / 16)   // 3125, exact

typedef __attribute__((ext_vector_type(2))) float v2f;
typedef __attribute__((ext_vector_type(4))) float v4f;
typedef __attribute__((ext_vector_type(8))) float v8f;

// ---------------------------------------------------------------- gather x
__global__ void gather_nodes(const int* __restrict__ xidx,
                             const float* __restrict__ emb,
                             float* __restrict__ x) {
    int t = blockIdx.x * blockDim.x + threadIdx.x;   // NNODES*32 threads
    int node = t >> 5, lane = t & 31;
    if (node < NNODES) {
        v4f v = *(const v4f*)(emb + xidx[node] * HID + lane * 4);
        *(v4f*)(x + (size_t)node * HID + lane * 4) = v;
    }
}

// ------------------------------------------------- edge messages + scatter
// one wave per edge; e_emb indexed on the fly (4x128 table, L0-resident)
__global__ void edge_msg(const float* __restrict__ x,
                         const int* __restrict__ src,
                         const int* __restrict__ dst,
                         const int* __restrict__ attr,
                         const float* __restrict__ eemb,
                         float* __restrict__ aggr) {
    int t = blockIdx.x * blockDim.x + threadIdx.x;   // NEDGES*32 threads
    int e = t >> 5, lane = t & 31;
    if (e < NEDGES) {
        int s = src[e], d = dst[e], a = attr[e];     // wave-uniform -> scalar
        int c = lane * 4;
        v4f xv = *(const v4f*)(x + (size_t)s * HID + c);
        v4f ev = *(const v4f*)(eemb + a * HID + c);
        float* out = aggr + (size_t)d * HID + c;
#pragma unroll
        for (int i = 0; i < 4; ++i) {
            float m = xv[i] + ev[i];
            m = m > 0.f ? m : 0.f;
            unsafeAtomicAdd(out + i, m);             // global_atomic_add_f32
        }
    }
}

// ------------------------------------------------------ fused MLP (WMMA)
// block = 256 thr = 8 waves; wave w owns output columns [16w,16w+16).
// W1/W2 column panels live in registers (loop-invariant), A tile in LDS.
// __launch_bounds__(256, 1): max per-wave VGPR budget -> no weight spills.
__global__ void __launch_bounds__(256, 1)
mlp_wmma(const float* __restrict__ hin,
         const float* __restrict__ W1, const float* __restrict__ b1,
         const float* __restrict__ W2, const float* __restrict__ b2,
         float* __restrict__ hout) {
    __shared__ float lA[16 * HID];   // 8 KB input tile
    __shared__ float lH[16 * HID];   // 8 KB hidden tile

    const int lane = threadIdx.x & 31;
    const int wave = threadIdx.x >> 5;
    const int n0   = wave * 16;
    const int n    = lane & 15;          // output column within tile
    const int m    = lane & 15;          // A-row this lane feeds
    const int hi   = lane >> 4;          // 0: K+{0,1}, 1: K+{2,3}
    const int kb   = hi * 2;

    // preload B panels: w[k] holds W[4k+kb][col], W[4k+kb+1][col]
    v2f w1r[32], w2r[32];
#pragma unroll
    for (int k = 0; k < 32; ++k) {
        int kk = 4 * k + kb;
        w1r[k].x = W1[(kk    ) * HID + n0 + n];
        w1r[k].y = W1[(kk + 1) * HID + n0 + n];
        w2r[k].x = W2[(kk    ) * HID + n0 + n];
        w2r[k].y = W2[(kk + 1) * HID + n0 + n];
    }
    const float bias1 = b1[n0 + n];
    const float bias2 = b2[n0 + n];

    for (int rb = blockIdx.x; rb < ROWBLKS; rb += gridDim.x) {
        const int row0 = rb * 16;

        __syncthreads();                                    // lA safe to overwrite
        for (int i = threadIdx.x; i < 512; i += 256)        // 2048 floats, float4
            ((v4f*)lA)[i] = ((const v4f*)(hin + (size_t)row0 * HID))[i];
        __syncthreads();

        // GEMM1: [16x128] @ [128x16]
        v8f acc = {0.f,0.f,0.f,0.f,0.f,0.f,0.f,0.f};
#pragma unroll
        for (int k = 0; k < 32; ++k) {
            int kk = 4 * k + kb;
            v2f a; a.x = lA[m * HID + kk]; a.y = lA[m * HID + kk + 1];
            acc = __builtin_amdgcn_wmma_f32_16x16x4_f32(
                false, a, false, w1r[k], (short)0, acc, false, false);
        }
        // bias + relu, stage h1 tile to LDS
#pragma unroll
        for (int r = 0; r < 8; ++r) {
            int mm = r + 8 * hi;
            lH[mm * HID + n0 + n] = fmaxf(acc[r] + bias1, 0.f);
        }
        __syncthreads();

        // GEMM2: [16x128] @ [128x16]
        v8f acc2 = {0.f,0.f,0.f,0.f,0.f,0.f,0.f,0.f};
#pragma unroll
        for (int k = 0; k < 32; ++k) {
            int kk = 4 * k + kb;
            v2f a; a.x = lH[m * HID + kk]; a.y = lH[m * HID + kk + 1];
            acc2 = __builtin_amdgcn_wmma_f32_16x16x4_f32(
                false, a, false, w2r[k], (short)0, acc2, false, false);
        }
        // bias + relu, store h2
#pragma unroll
        for (int r = 0; r < 8; ++r) {
            int mm = r + 8 * hi;
            hout[(size_t)(row0 + mm) * HID + n0 + n] = fmaxf(acc2[r] + bias2, 0.f);
        }
    }
}

// ------------------------------------------------------------- BN stats
__global__ void bn_stats(const float* __restrict__ h, float* __restrict__ accum) {
    int c = threadIdx.x;                 // 128 channels
    float s = 0.f, q = 0.f;
    for (int i = blockIdx.x; i < NNODES; i += gridDim.x) {
        float v = h[(size_t)i * HID + c];
        s += v; q += v * v;
    }
    unsafeAtomicAdd(&accum[c], s);
    unsafeAtomicAdd(&accum[HID + c], q);
}

// ------------------------------------------------------ BN normalize -> x
__global__ void bn_norm(const float* __restrict__ h, const float* __restrict__ accum,
                        const float* __restrict__ g, const float* __restrict__ beta,
                        float* __restrict__ xout) {
    int c = threadIdx.x;
    float mu  = accum[c] * (1.f / NNODES);
    float var = accum[HID + c] * (1.f / NNODES) - mu * mu;
    float sc  = g[c] * rsqrtf(var + BN_EPS);
    float bt  = beta[c];
    for (int i = blockIdx.x; i < NNODES; i += gridDim.x)
        xout[(size_t)i * HID + c] = (h[(size_t)i * HID + c] - mu) * sc + bt;
}

// ----------------------------------------------------------------- utils
__global__ void zero_f32(float* __restrict__ p, int nelem) {
    int t = blockIdx.x * blockDim.x + threadIdx.x;
    if (t < nelem) p[t] = 0.f;
}

__global__ void pool_nodes(const float* __restrict__ x, const int* __restrict__ batch,
                           float* __restrict__ out) {
    int t = blockIdx.x * blockDim.x + threadIdx.x;   // NNODES*32 threads
    int node = t >> 5, lane = t & 31;
    if (node < NNODES) {
        int gidx = batch[node];
        int c = lane * 4;
        v4f v = *(const v4f*)(x + (size_t)node * HID + c);
#pragma unroll
        for (int i = 0; i < 4; ++i)
            unsafeAtomicAdd(out + gidx * HID + c + i, v[i]);
    }
}

// ----------------------------------------------------------------- launch
extern "C" void kernel_launch(void* const* d_in, const int* in_sizes, int n_in,
                              void* d_out, int out_size, void* d_ws, size_t ws_size,
                              hipStream_t stream) {
    const int*   x_idx    = (const int*)  d_in[0];
    const int*   eidx     = (const int*)  d_in[1];
    const int*   eattr    = (const int*)  d_in[2];
    const int*   batch    = (const int*)  d_in[3];
    const float* node_emb = (const float*)d_in[4];
    const float* edge_emb = (const float*)d_in[5];
    const float* W1       = (const float*)d_in[6];
    const float* b1       = (const float*)d_in[7];
    const float* W2       = (const float*)d_in[8];
    const float* b2       = (const float*)d_in[9];
    const float* bn_g     = (const float*)d_in[10];
    const float* bn_b     = (const float*)d_in[11];
    float* out = (float*)d_out;

    const size_t NH = (size_t)NNODES * HID;
    float* xbuf  = (float*)d_ws;        // current node features
    float* abuf  = xbuf + NH;           // x + sum(messages)
    float* hbuf  = abuf + NH;           // MLP output
    float* accum = hbuf + NH;           // 256 floats: sum | sumsq

    const int* src = eidx;
    const int* dst = eidx + NEDGES;

    gather_nodes<<<(NNODES * 32 + 255) / 256, 256, 0, stream>>>(x_idx, node_emb, xbuf);

    for (int l = 0; l < NLAYERS; ++l) {
        // aggr := x  (GINE eps=0 self term), then scatter-add messages
        hipMemcpyAsync(abuf, xbuf, NH * sizeof(float), hipMemcpyDeviceToDevice, stream);
        edge_msg<<<(NEDGES * 32 + 255) / 256, 256, 0, stream>>>(
            xbuf, src, dst, eattr, edge_emb, abuf);
        mlp_wmma<<<1024, 256, 0, stream>>>(
            abuf, W1 + l * HID * HID, b1 + l * HID,
                  W2 + l * HID * HID, b2 + l * HID, hbuf);
        zero_f32<<<1, 256, 0, stream>>>(accum, 2 * HID);
        bn_stats<<<1024, HID, 0, stream>>>(hbuf, accum);
        bn_norm<<<2048, HID, 0, stream>>>(hbuf, accum, bn_g + l * HID, bn_b + l * HID, xbuf);
    }

    zero_f32<<<(NGRAPHS * HID + 255) / 256, 256, 0, stream>>>(out, NGRAPHS * HID);
    pool_nodes<<<(NNODES * 32 + 255) / 256, 256, 0, stream>>>(xbuf, batch, out);
}